// LocalEncoder_59983513256112
// MI455X (gfx1250) — compile-verified
//
#include <hip/hip_runtime.h>

typedef __attribute__((ext_vector_type(2))) float v2f;
typedef __attribute__((ext_vector_type(8))) float v8f;

#define HID 128
#define HEADS 4
#define CPH 128
#define ED 16
#define NDIM 64
#define NLAYER 2
#define NEG_SLOPE 0.2f
#define BN_EPS 1e-5f

#define AS1 __attribute__((address_space(1)))
#define AS3 __attribute__((address_space(3)))

// matches clang builtin signature "vV4i*1V4i*3IiIi" (leaked by round-2 diagnostic)
typedef int v4i_b __attribute__((vector_size(4 * sizeof(int))));
typedef AS1 v4i_b* gv4i_ptr;
typedef AS3 v4i_b* lv4i_ptr;

#if __has_builtin(__builtin_amdgcn_global_load_async_to_lds_b128)
#define HAVE_ASYNC_LDS 1
#else
#define HAVE_ASYNC_LDS 0
#endif

__device__ __forceinline__ void wait_async_zero() {
#if __has_builtin(__builtin_amdgcn_s_wait_asynccnt)
    __builtin_amdgcn_s_wait_asynccnt(0);
#else
    asm volatile("s_wait_asynccnt 0x0" ::: "memory");
#endif
}

// ---------- helpers: order-preserving float<->uint for atomic max ----------
__device__ __forceinline__ unsigned f2ord(float f) {
    unsigned u = __float_as_uint(f);
    return (int)u >= 0 ? (u | 0x80000000u) : ~u;
}
__device__ __forceinline__ float ord2f(unsigned u) {
    return (u & 0x80000000u) ? __uint_as_float(u & 0x7FFFFFFFu) : __uint_as_float(~u);
}

// ---------- generic fills ----------
__global__ void fill_f32_kernel(float* p, float v, int n) {
    int i = blockIdx.x * blockDim.x + threadIdx.x;
    if (i < n) p[i] = v;
}
__global__ void fill_u32_kernel(unsigned* p, unsigned v, int n) {
    int i = blockIdx.x * blockDim.x + threadIdx.x;
    if (i < n) p[i] = v;
}

// ---------- WMMA f32 GEMM with async LDS staging of the A strip ----------
// D[M,Nc] = A[M,K] * B[K,Nc] (+bias, opt relu).
// grid = (M/16, Nc/64); block = 128 threads = 4 waves, one 16x16 tile per wave.
// A strip [16 x K] staged once per block into LDS via gfx1250 async global->LDS
// DMA (ASYNCcnt tracked); row pitch K+4 dwords -> 16B-aligned rows and
// conflict-free ds_load_b64 reads (bank = (4*lane + k) % 64).
// WMMA f32 16x16x4 lane layout per CDNA5 ISA:
//   A: lanes 0-15 K={0,1}, lanes 16-31 K={2,3};  B mirrors A over rows;
//   C: VGPR r holds M=r (lanes 0-15) / M=r+8 (lanes 16-31), N=lane&15.
template <int K>
__global__ __launch_bounds__(128)
void wmma_gemm_lds_kernel(const float* __restrict__ A, const float* __restrict__ B,
                          const float* __restrict__ bias, float* __restrict__ D,
                          int M, int Nc, int relu_flag)
{
    constexpr int LDK = K + 4;                 // padded row pitch (dwords)
    constexpr int KQ  = K >> 2;                // float4 chunks per row
    __shared__ float A_lds[16 * LDK];

    const int tm   = blockIdx.x;
    const int tn   = blockIdx.y * 4 + (threadIdx.x >> 5);
    const int lane = threadIdx.x & 31;
    const int half = lane >> 4;
    const int l16  = lane & 15;

    // ---- stage A strip [16 x K] into LDS ----
#pragma unroll
    for (int c = threadIdx.x; c < 16 * KQ; c += 128) {
        const int r  = c / KQ;
        const int k4 = (c - r * KQ) << 2;
        const float* gp = A + (size_t)(tm * 16 + r) * K + k4;
        float* lp = &A_lds[r * LDK + k4];
#if HAVE_ASYNC_LDS
        __builtin_amdgcn_global_load_async_to_lds_b128((gv4i_ptr)gp, (lv4i_ptr)lp, 0, 0);
#else
        lp[0] = gp[0]; lp[1] = gp[1]; lp[2] = gp[2]; lp[3] = gp[3];
#endif
    }
#if HAVE_ASYNC_LDS
    wait_async_zero();
#endif
    __syncthreads();

    // ---- WMMA main loop: A from LDS, B from global (L2-resident weights) ----
    const float* bcol = B + tn * 16 + l16;
    const float* arow = &A_lds[l16 * LDK];
    v8f acc = {};
#pragma unroll 4
    for (int kk = 0; kk < K; kk += 4) {
        const int ka = kk + half * 2;
        v2f a, b;
        a.x = arow[ka];
        a.y = arow[ka + 1];
        b.x = bcol[(size_t)ka * Nc];
        b.y = bcol[(size_t)(ka + 1) * Nc];
        acc = __builtin_amdgcn_wmma_f32_16x16x4_f32(false, a, false, b,
                                                    (short)0, acc, false, false);
    }
    const int col = tn * 16 + l16;
    const float bv = bias ? bias[col] : 0.0f;
#pragma unroll
    for (int r = 0; r < 8; ++r) {
        float v = acc[r] + bv;
        if (relu_flag) v = fmaxf(v, 0.0f);
        D[(size_t)(tm * 16 + r + 8 * half) * Nc + col] = v;
    }
}

// ---------- precompute: column-sum of edge_attr (for self-loop 'mean' fill) ----------
__global__ void reduce_edge_attr_kernel(const float* __restrict__ ea, float* mea_sum, int E)
{
    int t = blockIdx.x * blockDim.x + threadIdx.x;
    int stride = gridDim.x * blockDim.x;
    float loc[ED];
#pragma unroll
    for (int j = 0; j < ED; ++j) loc[j] = 0.0f;
    for (int r = t; r < E; r += stride) {
        const float* row = ea + (size_t)r * ED;
#pragma unroll
        for (int j = 0; j < ED; ++j) loc[j] += row[j];
    }
#pragma unroll
    for (int j = 0; j < ED; ++j) atomicAdd(&mea_sum[j], loc[j]);
}

// ---------- precompute: P_src/P_dst/P_edge [L][HID][HEADS] ----------
__global__ void proj_kernel(const float* __restrict__ W_lin, const float* __restrict__ W_ledge,
                            const float* __restrict__ att_src, const float* __restrict__ att_dst,
                            const float* __restrict__ att_edge,
                            float* P_src, float* P_dst, float* P_edge)
{
    int idx = blockIdx.x * blockDim.x + threadIdx.x;
    const int total = NLAYER * 3 * HID * HEADS;
    if (idx >= total) return;
    int h = idx & (HEADS - 1);
    int tmp = idx >> 2;
    int k = tmp & (HID - 1);
    tmp >>= 7;
    int t = tmp % 3;
    int l = tmp / 3;
    const float* W   = (t == 2) ? W_ledge : W_lin;
    const float* att = (t == 0) ? att_src : (t == 1) ? att_dst : att_edge;
    const float* wrow = W + ((size_t)l * HID + k) * (HEADS * CPH) + (size_t)h * CPH;
    const float* arow = att + ((size_t)l * HEADS + h) * CPH;
    float s = 0.0f;
    for (int c = 0; c < CPH; ++c) s += wrow[c] * arow[c];
    float* P = (t == 0) ? P_src : (t == 1) ? P_dst : P_edge;
    P[((size_t)l * HID + k) * HEADS + h] = s;
}

// ---------- precompute: Q[L][ED][HEADS] = W_edge_enc @ P_edge ----------
__global__ void edgeq_kernel(const float* __restrict__ W_edge_enc,
                             const float* __restrict__ P_edge, float* Q)
{
    int idx = blockIdx.x * blockDim.x + threadIdx.x;   // L*ED*HEADS = 128
    if (idx >= NLAYER * ED * HEADS) return;
    int h = idx & (HEADS - 1);
    int d = (idx >> 2) & (ED - 1);
    int l = idx >> 6;
    float s = 0.0f;
    for (int k = 0; k < HID; ++k)
        s += W_edge_enc[(size_t)d * HID + k] * P_edge[((size_t)l * HID + k) * HEADS + h];
    Q[idx] = s;
}

// ---------- precompute: q0[L][HEADS], a_e for self loops ----------
__global__ void loopvals_kernel(const float* __restrict__ b_edge_enc,
                                const float* __restrict__ P_edge,
                                const float* __restrict__ Q,
                                const float* __restrict__ mea_sum, float invE,
                                float* q0v, float* aloop)
{
    int idx = threadIdx.x;                       // NLAYER*HEADS = 8 threads
    if (idx >= NLAYER * HEADS) return;
    int h = idx & (HEADS - 1);
    int l = idx >> 2;
    float q0 = 0.0f;
    for (int k = 0; k < HID; ++k)
        q0 += b_edge_enc[k] * P_edge[((size_t)l * HID + k) * HEADS + h];
    float a = q0;
    for (int d = 0; d < ED; ++d)
        a += (mea_sum[d] * invE) * Q[((size_t)l * ED + d) * HEADS + h];
    q0v[idx] = q0;
    aloop[idx] = a;
}

// ---------- node attention scores: a_s, a_d = h @ P ----------
__global__ void node_scores_kernel(const float* __restrict__ h,
                                   const float* __restrict__ Psrc_l,
                                   const float* __restrict__ Pdst_l,
                                   float* a_s, float* a_d, int N)
{
    int idx = blockIdx.x * blockDim.x + threadIdx.x;   // N * 8
    int n = idx >> 3;
    if (n >= N) return;
    int j = idx & 7;
    int hh = j & 3;
    const float* hr = h + (size_t)n * HID;
    const float* P = (j < 4) ? Psrc_l : Pdst_l;
    float s = 0.0f;
    for (int k = 0; k < HID; ++k) s += hr[k] * P[(size_t)k * HEADS + hh];
    if (j < 4) a_s[(size_t)n * HEADS + hh] = s;
    else       a_d[(size_t)n * HEADS + hh] = s;
}

// ---------- edge pass 1: alpha + segment-max (atomic on ordered uint) ----------
__global__ void edge_alpha_kernel(const int* __restrict__ src, const int* __restrict__ dst,
                                  const float* __restrict__ edge_attr,
                                  const float* __restrict__ Ql, const float* __restrict__ q0l,
                                  const float* __restrict__ aloopl,
                                  const float* __restrict__ a_s, const float* __restrict__ a_d,
                                  float* alpha, unsigned* amaxk, int E, int N)
{
    int e = blockIdx.x * blockDim.x + threadIdx.x;
    int E2 = E + N;
    if (e >= E2) return;
    int s, d;
    float ae[HEADS];
    if (e < E) {
        s = src[e]; d = dst[e];
        float ea[ED];
#pragma unroll
        for (int j = 0; j < ED; ++j) ea[j] = edge_attr[(size_t)e * ED + j];
#pragma unroll
        for (int h = 0; h < HEADS; ++h) {
            float t = q0l[h];
#pragma unroll
            for (int j = 0; j < ED; ++j) t += ea[j] * Ql[j * HEADS + h];
            ae[h] = t;
        }
    } else {
        s = d = e - E;
#pragma unroll
        for (int h = 0; h < HEADS; ++h) ae[h] = aloopl[h];
    }
#pragma unroll
    for (int h = 0; h < HEADS; ++h) {
        float a = a_s[(size_t)s * HEADS + h] + a_d[(size_t)d * HEADS + h] + ae[h];
        a = (a > 0.0f) ? a : NEG_SLOPE * a;
        alpha[(size_t)e * HEADS + h] = a;
        atomicMax(&amaxk[(size_t)d * HEADS + h], f2ord(a));
    }
}

// ---------- edge pass 2: exp(alpha - max) and segment-sum denominator ----------
__global__ void edge_exp_kernel(const int* __restrict__ dst, float* alpha,
                                const unsigned* __restrict__ amaxk, float* denom,
                                int E, int N)
{
    int e = blockIdx.x * blockDim.x + threadIdx.x;
    int E2 = E + N;
    if (e >= E2) return;
    int d = (e < E) ? dst[e] : e - E;
#pragma unroll
    for (int h = 0; h < HEADS; ++h) {
        float m = ord2f(amaxk[(size_t)d * HEADS + h]);
        float ex = expf(alpha[(size_t)e * HEADS + h] - m);
        alpha[(size_t)e * HEADS + h] = ex;
        atomicAdd(&denom[(size_t)d * HEADS + h], ex);
    }
}

// ---------- edge pass 3: weighted scatter, head-mean fused (one wave/edge) ----------
__global__ __launch_bounds__(256)
void edge_aggregate_kernel(const int* __restrict__ src, const int* __restrict__ dst,
                           const float* __restrict__ ex, const float* __restrict__ denom,
                           const float* __restrict__ xs, float* acc, int E, int N)
{
    int w = (blockIdx.x * blockDim.x + threadIdx.x) >> 5;
    int E2 = E + N;
    if (w >= E2) return;
    int lane = threadIdx.x & 31;
    int s, d;
    if (w < E) { s = src[w]; d = dst[w]; } else { s = d = w - E; }
    float wt[HEADS];
#pragma unroll
    for (int h = 0; h < HEADS; ++h)
        wt[h] = ex[(size_t)w * HEADS + h] /
                (denom[(size_t)d * HEADS + h] + 1e-16f) * (1.0f / HEADS);
    const int c0 = lane * 4;                      // 4 channels per lane, 128 total
    float4 m = make_float4(0.f, 0.f, 0.f, 0.f);
#pragma unroll
    for (int h = 0; h < HEADS; ++h) {
        const float4 v = *(const float4*)(xs + (size_t)s * (HEADS * CPH) + h * CPH + c0);
        m.x += wt[h] * v.x; m.y += wt[h] * v.y; m.z += wt[h] * v.z; m.w += wt[h] * v.w;
    }
    float* o = acc + (size_t)d * HID + c0;
    atomicAdd(o + 0, m.x); atomicAdd(o + 1, m.y);
    atomicAdd(o + 2, m.z); atomicAdd(o + 3, m.w);
}

// ---------- batchnorm stats: per-channel sum / sumsq ----------
__global__ __launch_bounds__(HID)
void bn_stats_kernel(const float* __restrict__ acc, const float* __restrict__ bias_l,
                     float* sums, float* sumsq, int N)
{
    int c = threadIdx.x;                          // 128 threads = 1 channel each
    float s = 0.0f, q = 0.0f;
    float bv = bias_l[c];
    for (int n = blockIdx.x; n < N; n += gridDim.x) {
        float v = acc[(size_t)n * HID + c] + bv;
        s += v; q += v * v;
    }
    atomicAdd(&sums[c], s);
    atomicAdd(&sumsq[c], q);
}

// ---------- fused BN + ELU + residual ----------
__global__ void finalize_kernel(const float* __restrict__ acc, const float* __restrict__ bias_l,
                                const float* __restrict__ gamma_l, const float* __restrict__ beta_l,
                                const float* __restrict__ sums, const float* __restrict__ sumsq,
                                const float* __restrict__ hres, float* hout, int N)
{
    int i = blockIdx.x * blockDim.x + threadIdx.x;
    if (i >= N * HID) return;
    int c = i & (HID - 1);
    float invN = 1.0f / (float)N;
    float mu = sums[c] * invN;
    float var = sumsq[c] * invN - mu * mu;
    float v = acc[i] + bias_l[c];
    v = (v - mu) * rsqrtf(var + BN_EPS) * gamma_l[c] + beta_l[c];
    v = (v > 0.0f) ? v : (expf(v) - 1.0f);        // ELU
    hout[i] = v + hres[i];
}

// =========================================================================
extern "C" void kernel_launch(void* const* d_in, const int* in_sizes, int n_in,
                              void* d_out, int out_size, void* d_ws, size_t ws_size,
                              hipStream_t stream)
{
    const float* x          = (const float*)d_in[0];
    const int*   eidx       = (const int*)  d_in[1];
    const float* edge_attr  = (const float*)d_in[2];
    const float* W_node     = (const float*)d_in[3];
    const float* b_node     = (const float*)d_in[4];
    const float* W_edge_enc = (const float*)d_in[5];
    const float* b_edge_enc = (const float*)d_in[6];
    const float* W_lin      = (const float*)d_in[7];
    const float* W_ledge    = (const float*)d_in[8];
    const float* att_src    = (const float*)d_in[9];
    const float* att_dst    = (const float*)d_in[10];
    const float* att_edge   = (const float*)d_in[11];
    const float* bias       = (const float*)d_in[12];
    const float* bn_gamma   = (const float*)d_in[13];
    const float* bn_beta    = (const float*)d_in[14];

    const int N  = in_sizes[0] / NDIM;          // 30000 (multiple of 16)
    const int E  = in_sizes[1] / 2;             // 200000
    const int E2 = E + N;
    const int* src = eidx;
    const int* dst = eidx + E;

    // --- carve workspace ---
    char* ws = (char*)d_ws;
    size_t off = 0;
    auto carve = [&](size_t bytes) -> void* {
        void* p = ws + off;
        off += (bytes + 255) & ~(size_t)255;
        return p;
    };
    float*    h_cur  = (float*)   carve((size_t)N * HID * 4);       // h (residual input)
    float*    h_acc  = (float*)   carve((size_t)N * HID * 4);       // aggregation accumulator
    float*    xs     = (float*)   carve((size_t)N * HEADS * CPH * 4);
    float*    a_s    = (float*)   carve((size_t)N * HEADS * 4);
    float*    a_d    = (float*)   carve((size_t)N * HEADS * 4);
    float*    alpha  = (float*)   carve((size_t)E2 * HEADS * 4);
    unsigned* amaxk  = (unsigned*)carve((size_t)N * HEADS * 4);
    float*    denom  = (float*)   carve((size_t)N * HEADS * 4);
    float*    mea    = (float*)   carve(ED * 4);
    float*    P_src  = (float*)   carve((size_t)NLAYER * HID * HEADS * 4);
    float*    P_dst  = (float*)   carve((size_t)NLAYER * HID * HEADS * 4);
    float*    P_edge = (float*)   carve((size_t)NLAYER * HID * HEADS * 4);
    float*    Q      = (float*)   carve((size_t)NLAYER * ED * HEADS * 4);
    float*    q0v    = (float*)   carve((size_t)NLAYER * HEADS * 4);
    float*    aloop  = (float*)   carve((size_t)NLAYER * HEADS * 4);
    float*    bn_sum = (float*)   carve(HID * 4);
    float*    bn_sq  = (float*)   carve(HID * 4);
    (void)ws_size; (void)n_in;

    const int B = 256;
    auto blocks = [](long n, int b) { return (int)((n + b - 1) / b); };

    // --- precompute attention projections (algebraic folding of huge GEMMs) ---
    fill_f32_kernel<<<1, 64, 0, stream>>>(mea, 0.0f, ED);
    reduce_edge_attr_kernel<<<64, B, 0, stream>>>(edge_attr, mea, E);
    proj_kernel<<<blocks(NLAYER * 3 * HID * HEADS, B), B, 0, stream>>>(
        W_lin, W_ledge, att_src, att_dst, att_edge, P_src, P_dst, P_edge);
    edgeq_kernel<<<1, 128, 0, stream>>>(W_edge_enc, P_edge, Q);
    loopvals_kernel<<<1, 32, 0, stream>>>(b_edge_enc, P_edge, Q, mea,
                                          1.0f / (float)E, q0v, aloop);

    // --- node encoder: h = relu(x @ W_node + b_node), WMMA f32, K=64 ---
    {
        dim3 grid(N / 16, HID / 64);
        wmma_gemm_lds_kernel<NDIM><<<grid, 128, 0, stream>>>(
            x, W_node, b_node, h_cur, N, HID, 1);
    }

    for (int l = 0; l < NLAYER; ++l) {
        // xs = h @ W_lin[l]  [N,512], WMMA f32, K=128
        {
            dim3 grid(N / 16, (HEADS * CPH) / 64);
            wmma_gemm_lds_kernel<HID><<<grid, 128, 0, stream>>>(
                h_cur, W_lin + (size_t)l * HID * HEADS * CPH, nullptr, xs,
                N, HEADS * CPH, 0);
        }

        node_scores_kernel<<<blocks((long)N * 8, B), B, 0, stream>>>(
            h_cur, P_src + (size_t)l * HID * HEADS, P_dst + (size_t)l * HID * HEADS,
            a_s, a_d, N);

        fill_u32_kernel<<<blocks((long)N * HEADS, B), B, 0, stream>>>(amaxk, 0u, N * HEADS);
        fill_f32_kernel<<<blocks((long)N * HEADS, B), B, 0, stream>>>(denom, 0.0f, N * HEADS);
        fill_f32_kernel<<<blocks((long)N * HID, B), B, 0, stream>>>(h_acc, 0.0f, N * HID);

        edge_alpha_kernel<<<blocks(E2, B), B, 0, stream>>>(
            src, dst, edge_attr,
            Q + (size_t)l * ED * HEADS, q0v + (size_t)l * HEADS, aloop + (size_t)l * HEADS,
            a_s, a_d, alpha, amaxk, E, N);

        edge_exp_kernel<<<blocks(E2, B), B, 0, stream>>>(dst, alpha, amaxk, denom, E, N);

        edge_aggregate_kernel<<<blocks((long)E2 * 32, B), B, 0, stream>>>(
            src, dst, alpha, denom, xs, h_acc, E, N);

        fill_f32_kernel<<<1, HID, 0, stream>>>(bn_sum, 0.0f, HID);
        fill_f32_kernel<<<1, HID, 0, stream>>>(bn_sq, 0.0f, HID);
        bn_stats_kernel<<<256, HID, 0, stream>>>(h_acc, bias + (size_t)l * HID,
                                                 bn_sum, bn_sq, N);

        float* dest = (l == NLAYER - 1) ? (float*)d_out : h_cur;  // in-place residual update
        finalize_kernel<<<blocks((long)N * HID, B), B, 0, stream>>>(
            h_acc, bias + (size_t)l * HID, bn_gamma + (size_t)l * HID,
            bn_beta + (size_t)l * HID, bn_sum, bn_sq, h_cur, dest, N);
    }
    (void)out_size;
}